// GCNLayer_19636590477404
// MI455X (gfx1250) — compile-verified
//
#include <hip/hip_runtime.h>

typedef float v2f __attribute__((ext_vector_type(2)));
typedef float v8f __attribute__((ext_vector_type(8)));

// ---------------------------------------------------------------------------
// K1: zero the workspace region (deg_out, deg_in, agg are contiguous)
// ---------------------------------------------------------------------------
__global__ void gcn_zero_kernel(float* __restrict__ p, long long n) {
    long long i = (long long)blockIdx.x * blockDim.x + threadIdx.x;
    if (i < n) p[i] = 0.0f;
}

// ---------------------------------------------------------------------------
// K2: degree computation (segment-sum of ones) via float atomics
// ---------------------------------------------------------------------------
__global__ void gcn_degree_kernel(const int* __restrict__ src,
                                  const int* __restrict__ dst,
                                  float* __restrict__ deg_out,
                                  float* __restrict__ deg_in, int E) {
    int e = blockIdx.x * blockDim.x + threadIdx.x;
    if (e < E) {
        atomicAdd(deg_out + src[e], 1.0f);
        atomicAdd(deg_in  + dst[e], 1.0f);
    }
}

// ---------------------------------------------------------------------------
// K3: scatter-add  agg[dst] += feats[src] * rsqrt(max(deg_out[src],1))
// 16 threads per edge, float4 per thread (64 features).
// ---------------------------------------------------------------------------
__global__ void gcn_scatter_kernel(const float* __restrict__ feats,
                                   const int* __restrict__ src,
                                   const int* __restrict__ dst,
                                   const float* __restrict__ deg_out,
                                   float* __restrict__ agg, int E) {
    long long i = (long long)blockIdx.x * blockDim.x + threadIdx.x;
    long long total = (long long)E * 16;
    if (i >= total) return;
    int e  = (int)(i >> 4);
    int f0 = (int)(i & 15) * 4;
    int s = src[e];
    int d = dst[e];
    float sn = rsqrtf(fmaxf(deg_out[s], 1.0f));
    float4 v = *(const float4*)(feats + (size_t)s * 64 + f0);
    float* ap = agg + (size_t)d * 64 + f0;
    atomicAdd(ap + 0, v.x * sn);
    atomicAdd(ap + 1, v.y * sn);
    atomicAdd(ap + 2, v.z * sn);
    atomicAdd(ap + 3, v.w * sn);
}

// ---------------------------------------------------------------------------
// K4: fused  out = LN( relu( (agg*rsqrt(deg_in)) @ W + b ) + feats )
// One wave (32 lanes) computes a 16-row tile with V_WMMA_F32_16X16X4_F32.
// 8 waves / block. W cached in LDS; D tile staged in padded LDS for the
// per-row LayerNorm.
// ---------------------------------------------------------------------------
#define STG 66  // padded row stride (words) -> bank stride 2, conflict-free

__global__ __launch_bounds__(256)
void gcn_gemm_ln_kernel(const float* __restrict__ agg,
                        const float* __restrict__ deg_in,
                        const float* __restrict__ feats,
                        const float* __restrict__ W,
                        const float* __restrict__ bias,
                        const float* __restrict__ gamma,
                        const float* __restrict__ beta,
                        float* __restrict__ out,
                        int n_tiles, int N) {
    __shared__ float lW[64 * 64];          // 16 KB
    __shared__ float stage[8][16 * STG];   // 33 KB: per-wave 16x64 D tile

    const int tid  = threadIdx.x;
    const int wave = tid >> 5;
    const int lane = tid & 31;

    // cooperative load of W into LDS
    for (int i = tid; i < 64 * 64; i += 256) lW[i] = W[i];
    __syncthreads();

    const int tile = blockIdx.x * 8 + wave;
    if (tile >= n_tiles) return;

    const int row0 = tile * 16;
    const int mrow = lane & 15;        // A-matrix row this lane serves
    const int koff = (lane >> 4) * 2;  // K sub-offset: lanes 16-31 hold K+2,K+3
    const int j    = lane & 15;        // B/C column within 16-wide tile

    int arow_i = row0 + mrow;
    if (arow_i >= N) arow_i = N - 1;   // safe clamp (N%16==0 in practice)
    const float rnorm = rsqrtf(fmaxf(deg_in[arow_i], 1.0f));
    const float* __restrict__ arow = agg + (size_t)arow_i * 64;

    v8f c0 = {}, c1 = {}, c2 = {}, c3 = {};

    #pragma unroll
    for (int k = 0; k < 16; ++k) {      // K = 64 in steps of 4
        const int kk = 4 * k + koff;
        // A fragment (16x4 f32): lane holds A[mrow][kk], A[mrow][kk+1]
        v2f a;
        a.x = arow[kk]     * rnorm;
        a.y = arow[kk + 1] * rnorm;
        // B fragments (4x16 f32) for the four 16-col tiles
        const float* w0 = lW + kk * 64;
        const float* w1 = lW + (kk + 1) * 64;
        v2f b0, b1, b2, b3;
        b0.x = w0[j];      b0.y = w1[j];
        b1.x = w0[16 + j]; b1.y = w1[16 + j];
        b2.x = w0[32 + j]; b2.y = w1[32 + j];
        b3.x = w0[48 + j]; b3.y = w1[48 + j];

        c0 = __builtin_amdgcn_wmma_f32_16x16x4_f32(false, a, false, b0, (short)0, c0, false, false);
        c1 = __builtin_amdgcn_wmma_f32_16x16x4_f32(false, a, false, b1, (short)0, c1, false, false);
        c2 = __builtin_amdgcn_wmma_f32_16x16x4_f32(false, a, false, b2, (short)0, c2, false, false);
        c3 = __builtin_amdgcn_wmma_f32_16x16x4_f32(false, a, false, b3, (short)0, c3, false, false);
    }

    // Spill D tile to LDS: C/D layout -> row M = v + 8*(lane>=16), col = lane&15
    float* __restrict__ st = &stage[wave][0];
    const int half = (lane >> 4) * 8;
    #pragma unroll
    for (int v = 0; v < 8; ++v) {
        float* r = st + (half + v) * STG + j;
        r[0]  = c0[v];
        r[16] = c1[v];
        r[32] = c2[v];
        r[48] = c3[v];
    }
    __builtin_amdgcn_wave_barrier();   // same-wave DS ops are in-order; just
                                       // stop the compiler from reordering

    // Epilogue: 16 lanes, one row each: bias + relu + residual + LayerNorm
    if (lane < 16) {
        const int grow = row0 + lane;
        if (grow < N) {
            const float* __restrict__ frow = feats + (size_t)grow * 64;
            float* __restrict__ srow = st + lane * STG;
            float sum = 0.0f, sumsq = 0.0f;
            #pragma unroll 4
            for (int f = 0; f < 64; ++f) {
                float v = srow[f] + bias[f];
                v = fmaxf(v, 0.0f);        // relu (applied twice == once)
                v = v + frow[f];           // residual
                srow[f] = v;
                sum   += v;
                sumsq += v * v;
            }
            const float mu   = sum * (1.0f / 64.0f);
            const float var  = sumsq * (1.0f / 64.0f) - mu * mu;
            const float rstd = rsqrtf(var + 1e-5f);
            float* __restrict__ orow = out + (size_t)grow * 64;
            #pragma unroll 4
            for (int f = 0; f < 64; ++f) {
                orow[f] = (srow[f] - mu) * rstd * gamma[f] + beta[f];
            }
        }
    }
}

// ---------------------------------------------------------------------------
// launch
// ---------------------------------------------------------------------------
extern "C" void kernel_launch(void* const* d_in, const int* in_sizes, int n_in,
                              void* d_out, int out_size, void* d_ws, size_t ws_size,
                              hipStream_t stream) {
    const float* feats = (const float*)d_in[0];
    const int*   src   = (const int*)d_in[1];
    const int*   dst   = (const int*)d_in[2];
    const float* W     = (const float*)d_in[3];
    const float* bias  = (const float*)d_in[4];
    const float* gamma = (const float*)d_in[5];
    const float* beta  = (const float*)d_in[6];

    const int N = in_sizes[0] / 64;
    const int E = in_sizes[1];

    float* ws      = (float*)d_ws;
    float* deg_out = ws;                     // [N]
    float* deg_in  = ws + N;                 // [N]
    float* agg     = ws + 2 * (size_t)N;     // [N*64]

    // K1: zero deg_out | deg_in | agg (contiguous: N*66 floats)
    long long zn = (long long)N * 66;
    gcn_zero_kernel<<<(int)((zn + 255) / 256), 256, 0, stream>>>(ws, zn);

    // K2: degrees
    gcn_degree_kernel<<<(E + 255) / 256, 256, 0, stream>>>(src, dst, deg_out, deg_in, E);

    // K3: normalized scatter-add
    long long sc = (long long)E * 16;
    gcn_scatter_kernel<<<(int)((sc + 255) / 256), 256, 0, stream>>>(
        feats, src, dst, deg_out, agg, E);

    // K4: WMMA GEMM + bias + relu + residual + LayerNorm
    const int n_tiles = (N + 15) / 16;
    gcn_gemm_ln_kernel<<<(n_tiles + 7) / 8, 256, 0, stream>>>(
        agg, deg_in, feats, W, bias, gamma, beta, (float*)d_out, n_tiles, N);
}